// MoveRankingModel_5196910428205
// MI455X (gfx1250) — compile-verified
//
#include <hip/hip_runtime.h>
#include <stdint.h>

// ---- vector types for CDNA5 builtins ----
typedef float    v2f   __attribute__((ext_vector_type(2)));
typedef float    v8f   __attribute__((ext_vector_type(8)));
typedef unsigned u32x4 __attribute__((ext_vector_type(4)));
typedef int      i32x8 __attribute__((ext_vector_type(8)));
typedef int      i32x4 __attribute__((ext_vector_type(4)));

// Problem constants (from reference): B=16384, P=32, M=64, V=16, V2=8
#define P_OCC   32
#define M_MOV   64
#define V_DIM   16
#define V2_DIM  8
#define NEXP    384              // 6*64 experts
#define NCOL    (NEXP * V2_DIM)  // 3072 columns of W
#define NCT     (NCOL / 16)      // 192 column tiles
#define WT_STRIDE 3080           // 3072 + 8 pad -> conflict-free B-operand loads
#define BV_STRIDE 17             // 16 + 1 pad  -> conflict-free A-operand loads
#define SA_STRIDE 385            // 384 + 1 pad -> spread store banks
#define TB_BLK  64               // batches per block (4 waves x 16)
#define NTHR    128

// ---- LDS layout (bytes) ----
#define LDS_WT_OFF 0
#define LDS_WT_SZ  (V_DIM * WT_STRIDE * 4)        // 197120
#define LDS_BV_OFF (LDS_WT_OFF + LDS_WT_SZ)       // 197120
#define LDS_BV_SZ  (TB_BLK * BV_STRIDE * 4)       // 4352
#define LDS_SA_OFF (LDS_BV_OFF + LDS_BV_SZ)       // 201472
#define LDS_SA_SZ  (4 * 16 * SA_STRIDE * 4)       // 98560
#define LDS_B2_OFF (LDS_SA_OFF + LDS_SA_SZ)       // 300032
#define LDS_B2_SZ  (NCOL * 4)                     // 12288
#define LDS_OW_OFF (LDS_B2_OFF + LDS_B2_SZ)       // 312320
#define LDS_OW_SZ  (NCOL * 4)                     // 12288
#define LDS_OB_OFF (LDS_OW_OFF + LDS_OW_SZ)       // 324608
#define LDS_OB_SZ  (NEXP * 4)                     // 1536
#define LDS_TOTAL  (LDS_OB_OFF + LDS_OB_SZ)       // 326144 <= 327680 (320KB)

// 8-lane sum (groups {0-7},{8-15},{16-23},{24-31}) via DPP: quad_perm xor1,
// quad_perm xor2, row_half_mirror.  Every lane ends with its group's sum.
__device__ __forceinline__ float red8_dpp(float v) {
    int y;
    y = __builtin_amdgcn_update_dpp(0, __builtin_bit_cast(int, v), 0xB1, 0xF, 0xF, true);
    v += __builtin_bit_cast(float, y);
    y = __builtin_amdgcn_update_dpp(0, __builtin_bit_cast(int, v), 0x4E, 0xF, 0xF, true);
    v += __builtin_bit_cast(float, y);
    y = __builtin_amdgcn_update_dpp(0, __builtin_bit_cast(int, v), 0x141, 0xF, 0xF, true);
    v += __builtin_bit_cast(float, y);
    return v;
}

// One TDM descriptor issue (2D tile: nline lines of len0 contiguous elems,
// line stride lstride elems, 4-byte data).  1D copies pass nline=0.
__device__ __forceinline__ void tdm_load(uint32_t ldsa, uint64_t ga,
                                         uint32_t len0, uint32_t nline,
                                         uint32_t lstride) {
    u32x4 g0;
    g0[0] = 1u;                                   // count=1, user load-descriptor
    g0[1] = ldsa;                                 // lds_addr (bytes)
    g0[2] = (uint32_t)ga;                         // global_addr[31:0]
    g0[3] = (uint32_t)((ga >> 32) & 0x1FFFFFFu)   // global_addr[56:32]
          | (2u << 30);                           // type = 2 ("image")
    i32x8 g1;
    g1[0] = (int)(2u << 16);          // data_size=2 (4B); no mask/pad/iterate
    g1[1] = (int)(len0 << 16);        // tensor_dim0 low16 (bits 79:48)
    g1[2] = (int)(nline << 16);       // tensor_dim0 hi=0 | tensor_dim1 low16
    g1[3] = (int)(len0 << 16);        // tensor_dim1 hi=0 | tile_dim0
    g1[4] = (int)nline;               // tile_dim1 (0 => 1D) | tile_dim2=0
    g1[5] = (int)lstride;             // tensor_dim0_stride low32 (elems)
    g1[6] = 0;                        // stride0 hi | tensor_dim1_stride low
    g1[7] = 0;
    i32x4 g2 = {0, 0, 0, 0};          // <=2D: groups 2/3 unused
    i32x4 g3 = {0, 0, 0, 0};
    i32x8 g4 = {0, 0, 0, 0, 0, 0, 0, 0};
    __builtin_amdgcn_tensor_load_to_lds(g0, g1, g2, g3, g4, 0);
}

__global__ void __launch_bounds__(NTHR)
move_rank_wmma_kernel(const int* __restrict__ occ_idx,       // [B,32]
                      const int* __restrict__ move_idx,      // [B,64]
                      const float* __restrict__ psv,         // [768,16]
                      const float* __restrict__ psb,         // [16]
                      const float* __restrict__ mvw,         // [384,16,8]
                      const float* __restrict__ b2f,         // [3072]
                      const float* __restrict__ owf,         // [3072]
                      const float* __restrict__ obf,         // [384]
                      float* __restrict__ out)               // [B,64]
{
    extern __shared__ char smem[];
    float* wt_s  = (float*)(smem + LDS_WT_OFF);   // [16][3080]  W^T
    float* bvec_s= (float*)(smem + LDS_BV_OFF);   // [64][17]    board vectors
    float* sall_s= (float*)(smem + LDS_SA_OFF);   // [4][16][385] all-expert scores
    float* b2_s  = (float*)(smem + LDS_B2_OFF);   // [3072]
    float* ow_s  = (float*)(smem + LDS_OW_OFF);   // [3072]
    float* ob_s  = (float*)(smem + LDS_OB_OFF);   // [384]

    const int t   = threadIdx.x;
    const int bg0 = blockIdx.x * TB_BLK;

    // ---------------- Phase 0: TDM staging (wave 0 issues; EXEC ignored) ----
    // Wt[d][e*8+k] = mvw[e][d][k]: per d-row, 384 lines x 8 contiguous elems,
    // line stride 128 elems.  Plus 1D copies of b2 / ow / ob.
    if (t == 0) {
        const uint64_t gmv = (uint64_t)(uintptr_t)mvw;
        const uint32_t l0  = (uint32_t)(uintptr_t)smem;
        #pragma unroll 1
        for (int d = 0; d < V_DIM; ++d)
            tdm_load(l0 + LDS_WT_OFF + (uint32_t)d * (WT_STRIDE * 4),
                     gmv + (uint64_t)d * (V2_DIM * 4), 8, NEXP, 128);
        tdm_load(l0 + LDS_B2_OFF, (uint64_t)(uintptr_t)b2f, NCOL, 0, NCOL);
        tdm_load(l0 + LDS_OW_OFF, (uint64_t)(uintptr_t)owf, NCOL, 0, NCOL);
        tdm_load(l0 + LDS_OB_OFF, (uint64_t)(uintptr_t)obf, NEXP, 0, NEXP);
    }

    // ---------------- Phase A (overlaps TDM): board vectors ----------------
    int* occ_s = (int*)sall_s;                    // S_all region reused
    for (int i = t; i < TB_BLK * P_OCC; i += NTHR)
        occ_s[i] = occ_idx[bg0 * P_OCC + i];
    __syncthreads();

    for (int i = t; i < TB_BLK * V_DIM; i += NTHR) {
        int b = i >> 4, d = i & 15;
        float s = psb[d];
        #pragma unroll 8
        for (int p = 0; p < P_OCC; ++p)
            s += psv[occ_s[b * P_OCC + p] * V_DIM + d];
        bvec_s[b * BV_STRIDE + d] = s;
    }

    __builtin_amdgcn_s_wait_tensorcnt(0);
    __syncthreads();

    // ---------------- Phase B: dense all-expert WMMA scoring ---------------
    const int wv   = t >> 5;
    const int lane = t & 31;
    const int row  = lane & 15;
    const int hiK  = (lane >> 4) << 1;           // K offset 0 / 2 per half-wave
    const int sv   = lane & 7;                   // which v this lane stores
    const int rrow = sv + ((lane & 16) ? 8 : 0); // D row it stores
    const int eoff = (lane >> 3) & 1;            // expert within tile

    v2f A[4];
    #pragma unroll
    for (int kc = 0; kc < 4; ++kc) {
        const float* bp = &bvec_s[(wv * 16 + row) * BV_STRIDE + kc * 4 + hiK];
        A[kc].x = bp[0];
        A[kc].y = bp[1];
    }

    float* Sw = sall_s + wv * 16 * SA_STRIDE;

    // double-buffered B operand + per-column bias/weight
    v2f Bc[4], Bn[4];
    #pragma unroll
    for (int kc = 0; kc < 4; ++kc) {
        Bc[kc].x = wt_s[(kc * 4 + hiK) * WT_STRIDE + row];
        Bc[kc].y = wt_s[(kc * 4 + hiK + 1) * WT_STRIDE + row];
    }
    float b2v = b2_s[row], owv = ow_s[row];

    #pragma unroll 1
    for (int ct = 0; ct < NCT; ++ct) {
        v8f c = {};
        #pragma unroll
        for (int kc = 0; kc < 4; ++kc)
            c = __builtin_amdgcn_wmma_f32_16x16x4_f32(
                    false, A[kc], false, Bc[kc], (short)0, c, false, false);

        // prefetch next tile's operands while post-processing this one
        int ctn = (ct + 1 < NCT) ? (ct + 1) : (NCT - 1);
        int coln = ctn * 16 + row;
        #pragma unroll
        for (int kc = 0; kc < 4; ++kc) {
            Bn[kc].x = wt_s[(kc * 4 + hiK) * WT_STRIDE + coln];
            Bn[kc].y = wt_s[(kc * 4 + hiK + 1) * WT_STRIDE + coln];
        }
        float b2n = b2_s[coln], own = ow_s[coln];

        // relu(+b2) * ow, 8-lane DPP reduce per D-register
        float r0 = red8_dpp(fmaxf(c[0] + b2v, 0.0f) * owv);
        float r1 = red8_dpp(fmaxf(c[1] + b2v, 0.0f) * owv);
        float r2 = red8_dpp(fmaxf(c[2] + b2v, 0.0f) * owv);
        float r3 = red8_dpp(fmaxf(c[3] + b2v, 0.0f) * owv);
        float r4 = red8_dpp(fmaxf(c[4] + b2v, 0.0f) * owv);
        float r5 = red8_dpp(fmaxf(c[5] + b2v, 0.0f) * owv);
        float r6 = red8_dpp(fmaxf(c[6] + b2v, 0.0f) * owv);
        float r7 = red8_dpp(fmaxf(c[7] + b2v, 0.0f) * owv);

        // lane l stores v = l&7: 32 lanes -> 32 distinct (row, expert) slots
        float p01 = (sv & 1) ? r1 : r0;
        float p23 = (sv & 1) ? r3 : r2;
        float p45 = (sv & 1) ? r5 : r4;
        float p67 = (sv & 1) ? r7 : r6;
        float p03 = (sv & 2) ? p23 : p01;
        float p47 = (sv & 2) ? p67 : p45;
        float pck = (sv & 4) ? p47 : p03;
        Sw[rrow * SA_STRIDE + 2 * ct + eoff] = pck;

        #pragma unroll
        for (int kc = 0; kc < 4; ++kc) Bc[kc] = Bn[kc];
        b2v = b2n; owv = own;
    }
    __syncthreads();

    // ---------------- Phase C: gather scores -------------------------------
    for (int o = t; o < TB_BLK * M_MOV; o += NTHR) {
        int bl = o >> 6, m = o & 63;
        int e  = move_idx[(bg0 + bl) * M_MOV + m];
        float s = sall_s[(bl >> 4) * (16 * SA_STRIDE) + (bl & 15) * SA_STRIDE + e]
                + ob_s[e];
        out[(bg0 + bl) * M_MOV + m] = s;
    }
}

extern "C" void kernel_launch(void* const* d_in, const int* in_sizes, int n_in,
                              void* d_out, int out_size, void* d_ws, size_t ws_size,
                              hipStream_t stream) {
    const int*   occ_idx = (const int*)  d_in[0];
    const int*   mov_idx = (const int*)  d_in[1];
    const float* psv     = (const float*)d_in[2];
    const float* psb     = (const float*)d_in[3];
    const float* mvw     = (const float*)d_in[4];
    const float* b2f     = (const float*)d_in[5];
    const float* owf     = (const float*)d_in[6];
    const float* obf     = (const float*)d_in[7];
    float*       out     = (float*)d_out;

    const int B = in_sizes[0] / P_OCC;      // 16384
    const int grid = B / TB_BLK;            // 256 blocks

    move_rank_wmma_kernel<<<grid, NTHR, LDS_TOTAL, stream>>>(
        occ_idx, mov_idx, psv, psb, mvw, b2f, owf, obf, out);
}